// RelationDetrHungarianMatcher_54288386621697
// MI455X (gfx1250) — compile-verified
//
#include <hip/hip_runtime.h>
#include <cstdint>
#include <cstddef>

// Problem constants (from reference)
#define N_PRED 20000
#define M_GT   1000
#define MP     1024      // M padded to multiple of 256
#define NCLS   91
#define ROWS   64        // pred rows per block in the main kernel
#define TPB    256

// Workspace layout (bytes); all offsets 16B-aligned
#define WS_CT     0                           // N*C floats: focal class-cost table
#define WS_GT     (N_PRED * NCLS * 4)         // 6*MP floats: gt x1|y1|x2|y2|area|label*4
#define WS_PREDC  (WS_GT + 6 * MP * 4)        // N*4 floats: pred corners
#define WS_PREDA  (WS_PREDC + N_PRED * 4 * 4) // N floats: pred areas

// ---------------------------------------------------------------------------
// MAIN kernel (placed first so the disasm snippet shows it).
// Block = 256 threads, covers ROWS pred rows x all M columns. GT pack (24 KB)
// staged via async global->LDS (ASYNCcnt), then register-cached (4 columns
// per thread). Pred row data read via uniform (scalar) loads. Class cost
// gathered from the precomputed L2-resident table. Output stores are
// non-temporal (write-once stream, 80 MB).
// ---------------------------------------------------------------------------
__global__ __launch_bounds__(TPB) void cost_kernel(
    const float* __restrict__ predc, const float* __restrict__ preda,
    const float* __restrict__ ct, const float* __restrict__ gtpack,
    float* __restrict__ out) {
  __shared__ float lds[6 * MP];                 // 24 KB
  const int t = threadIdx.x;

  // LDS byte address of our buffer: low 32 bits of the generic pointer are
  // the LDS_BASE-relative address (ISA 10.2 aperture mapping).
  unsigned loff = (unsigned)(uintptr_t)(void*)&lds[0];

  // Stage 24 KB of GT pack: 1536 x 16B chunks, 6 per thread, async to LDS.
  const char* gsrc = (const char*)gtpack;
#pragma unroll
  for (int k = 0; k < 6; ++k) {
    unsigned chunk = (unsigned)(k * TPB + t);
    unsigned ldst  = loff + chunk * 16u;
    const char* g  = gsrc + (size_t)chunk * 16u;
    asm volatile("global_load_async_to_lds_b128 %0, %1, off"
                 :: "v"(ldst), "v"(g)
                 : "memory");
  }
  asm volatile("s_wait_asynccnt 0" ::: "memory");
  __syncthreads();

  // Register-cache this thread's 4 columns of GT state.
  float gx1[4], gy1[4], gx2[4], gy2[4], ga[4];
  int go[4];
#pragma unroll
  for (int jj = 0; jj < 4; ++jj) {
    int j = t + jj * TPB;                       // < MP, padded entries are 0
    gx1[jj] = lds[0 * MP + j];
    gy1[jj] = lds[1 * MP + j];
    gx2[jj] = lds[2 * MP + j];
    gy2[jj] = lds[3 * MP + j];
    ga[jj]  = lds[4 * MP + j];
    go[jj]  = ((const int*)lds)[5 * MP + j];
  }

  const int i0   = blockIdx.x * ROWS;
  const int iend = (i0 + ROWS < N_PRED) ? (i0 + ROWS) : N_PRED;

  for (int i = i0; i < iend; ++i) {
    // Uniform addresses -> scalar (SMEM) loads.
    float4 pc = *(const float4*)(predc + (size_t)i * 4);
    float  pa = preda[i];
    const char* ctrow = (const char*)ct + (size_t)i * (NCLS * 4);
    float* orow = out + (size_t)i * M_GT;

#pragma unroll
    for (int jj = 0; jj < 4; ++jj) {
      int j = t + jj * TPB;
      if (j < M_GT) {                           // folds to true for jj<3
        float cc = *(const float*)(ctrow + go[jj]);   // class-cost gather

        // IoU
        float ix1 = fmaxf(pc.x, gx1[jj]), iy1 = fmaxf(pc.y, gy1[jj]);
        float ix2 = fminf(pc.z, gx2[jj]), iy2 = fminf(pc.w, gy2[jj]);
        float iw  = fmaxf(ix2 - ix1, 0.0f), ih = fmaxf(iy2 - iy1, 0.0f);
        float inter = iw * ih;
        float uni   = pa + ga[jj] - inter;

        // Enclosing box (diffs are >= 0 by construction)
        float ex1 = fminf(pc.x, gx1[jj]), ey1 = fminf(pc.y, gy1[jj]);
        float ex2 = fmaxf(pc.z, gx2[jj]), ey2 = fmaxf(pc.w, gy2[jj]);
        float ea  = (ex2 - ex1) * (ey2 - ey1);

        float runi = __builtin_amdgcn_rcpf(uni);
        float rea  = __builtin_amdgcn_rcpf(ea);
        float giou = inter * runi - (ea - uni) * rea;

        // Exact cxcywh L1 from corner diffs:
        // |dcx|+|dw| = 0.5*|dx1+dx2| + |dx2-dx1| (same for y)
        float dx1 = pc.x - gx1[jj], dx2 = pc.z - gx2[jj];
        float dy1 = pc.y - gy1[jj], dy2 = pc.w - gy2[jj];
        float l1 = 0.5f * fabsf(dx1 + dx2) + fabsf(dx2 - dx1)
                 + 0.5f * fabsf(dy1 + dy2) + fabsf(dy2 - dy1);

        __builtin_nontemporal_store(cc + l1 - giou, &orow[j]);
      }
    }
  }
}

// ---------------------------------------------------------------------------
// Kernel 1: focal class-cost table ct[i][c] for all N x C logits.
// class_cost = ALPHA*(1-p)^2*softplus(-x) - (1-ALPHA)*p^2*softplus(x)
// ---------------------------------------------------------------------------
__global__ __launch_bounds__(TPB) void class_table_kernel(
    const float* __restrict__ logits, float* __restrict__ ct) {
  int idx = blockIdx.x * TPB + threadIdx.x;
  if (idx >= N_PRED * NCLS) return;
  float x  = logits[idx];
  float ax = fabsf(x);
  float e  = __expf(-ax);                       // exp(-|x|) in (0,1]
  float rc = __builtin_amdgcn_rcpf(1.0f + e);
  float lp = __logf(1.0f + e);                  // log1p(exp(-|x|)), arg in (1,2]
  float sp_pos = fmaxf(x, 0.0f) + lp;           // softplus(x)
  float sp_neg = sp_pos - x;                    // softplus(-x)
  float p = (x >= 0.0f ? 1.0f : e) * rc;        // sigmoid(x), stable
  float q = 1.0f - p;
  ct[idx] = 0.25f * q * q * sp_neg - 0.75f * p * p * sp_pos;
}

// ---------------------------------------------------------------------------
// Kernel 2: GT prep -> SoA pack [x1|y1|x2|y2|area|label*4], padded to MP=1024
// ---------------------------------------------------------------------------
__global__ __launch_bounds__(TPB) void gt_prep_kernel(
    const float* __restrict__ gt, const int* __restrict__ lab,
    float* __restrict__ gp) {
  int j = blockIdx.x * TPB + threadIdx.x;
  if (j >= MP) return;
  float cx = 0.f, cy = 0.f, w = 0.f, h = 0.f;
  int lb = 0;
  if (j < M_GT) {
    cx = gt[j * 4 + 0]; cy = gt[j * 4 + 1];
    w  = gt[j * 4 + 2]; h  = gt[j * 4 + 3];
    lb = lab[j];
  }
  gp[0 * MP + j] = cx - 0.5f * w;
  gp[1 * MP + j] = cy - 0.5f * h;
  gp[2 * MP + j] = cx + 0.5f * w;
  gp[3 * MP + j] = cy + 0.5f * h;
  gp[4 * MP + j] = w * h;
  ((int*)gp)[5 * MP + j] = lb * 4;              // byte offset into ct row
}

// ---------------------------------------------------------------------------
// Kernel 3: pred prep -> corners (AoS float4) + area
// ---------------------------------------------------------------------------
__global__ __launch_bounds__(TPB) void pred_prep_kernel(
    const float* __restrict__ pb, float* __restrict__ pc,
    float* __restrict__ pa) {
  int i = blockIdx.x * TPB + threadIdx.x;
  if (i >= N_PRED) return;
  float cx = pb[i * 4 + 0], cy = pb[i * 4 + 1];
  float w  = pb[i * 4 + 2], h  = pb[i * 4 + 3];
  pc[i * 4 + 0] = cx - 0.5f * w;
  pc[i * 4 + 1] = cy - 0.5f * h;
  pc[i * 4 + 2] = cx + 0.5f * w;
  pc[i * 4 + 3] = cy + 0.5f * h;
  pa[i] = w * h;
}

// ---------------------------------------------------------------------------
extern "C" void kernel_launch(void* const* d_in, const int* in_sizes, int n_in,
                              void* d_out, int out_size, void* d_ws,
                              size_t ws_size, hipStream_t stream) {
  const float* pred_boxes  = (const float*)d_in[0];   // N x 4 f32
  const float* pred_logits = (const float*)d_in[1];   // N x C f32
  const float* gt_boxes    = (const float*)d_in[2];   // M x 4 f32
  const int*   gt_labels   = (const int*)d_in[3];     // M   i32
  float* out = (float*)d_out;                         // N x M f32

  char* ws = (char*)d_ws;
  float* ct     = (float*)(ws + WS_CT);
  float* gtpack = (float*)(ws + WS_GT);
  float* predc  = (float*)(ws + WS_PREDC);
  float* preda  = (float*)(ws + WS_PREDA);

  class_table_kernel<<<(N_PRED * NCLS + TPB - 1) / TPB, TPB, 0, stream>>>(
      pred_logits, ct);
  gt_prep_kernel<<<MP / TPB, TPB, 0, stream>>>(gt_boxes, gt_labels, gtpack);
  pred_prep_kernel<<<(N_PRED + TPB - 1) / TPB, TPB, 0, stream>>>(
      pred_boxes, predc, preda);
  cost_kernel<<<(N_PRED + ROWS - 1) / ROWS, TPB, 0, stream>>>(
      predc, preda, ct, gtpack, out);
}